// NetAmazon_GCN_45148696215621
// MI455X (gfx1250) — compile-verified
//
#include <hip/hip_runtime.h>
#include <hip/hip_bf16.h>
#include <math.h>

// ---------------------------------------------------------------------------
// CDNA5 / gfx1250 GCN forward pass.
//   roofline: aggregation streams ~7 GB (gather + L2-resident scatter-adds)
//   -> ~300us at 23.3 TB/s.  GEMMs total <30 GFLOP -> f32 WMMA (16x16x4) is
//   free, so we keep full fp32 precision end-to-end.
//   All GEMM tiles are made fully in-bounds (W3 zero-padded to 64x16) so the
//   WMMA inner loop has no divergent predication: straight load+wmma.
// ---------------------------------------------------------------------------

typedef __attribute__((ext_vector_type(2))) float v2f;
typedef __attribute__((ext_vector_type(8))) float v8f;

// ---------------- degree / norm ----------------
__global__ void gcn_deg_init(float* deg, int n) {
    int i = blockIdx.x * blockDim.x + threadIdx.x;
    if (i < n) deg[i] = 1.0f;                      // self loop contributes 1
}

__global__ void gcn_deg_count(const int* __restrict__ dst, float* deg, int e) {
    int i = blockIdx.x * blockDim.x + threadIdx.x;
    if (i < e) unsafeAtomicAdd(&deg[dst[i]], 1.0f);
}

__global__ void gcn_deg_to_dinv(float* deg, int n) {
    int i = blockIdx.x * blockDim.x + threadIdx.x;
    if (i < n) deg[i] = rsqrtf(deg[i]);            // deg >= 1 always
}

// ---------------- zero-pad W3 (64x10 -> 64x16) ----------------
__global__ void gcn_pad_w3(const float* __restrict__ w3, float* __restrict__ out) {
    int i = blockIdx.x * blockDim.x + threadIdx.x;
    if (i >= 64 * 16) return;
    int k = i >> 4, n = i & 15;
    out[i] = (n < 10) ? w3[k * 10 + n] : 0.0f;
}

// ---------------- WMMA f32 GEMM:  C[M x (NT*16 per block.x)] = A * B --------
// One wave computes a 16(M) x NT*16(N) strip (A fragment reused NT times).
// 8 waves / block -> 128 rows per block.  All dimensions in-bounds by
// construction (N multiple of 16, M guarded at wave granularity so EXEC is
// all-ones at every WMMA).  No per-lane predication anywhere.
template <int NT>
__global__ void __launch_bounds__(256)
gcn_gemm_wmma(const float* __restrict__ A, const float* __restrict__ B,
              float* __restrict__ C, int M, int K,
              int lda, int ldb, int ldc) {
    const int wave = threadIdx.x >> 5;
    const int lane = threadIdx.x & 31;
    const int half = lane >> 4;       // 0: lanes 0-15, 1: lanes 16-31
    const int l    = lane & 15;

    const int mBase = (blockIdx.y * 8 + wave) * 16;
    if (mBase >= M) return;                         // wave-uniform exit
    const int nBase = blockIdx.x * (NT * 16);

    v8f acc[NT];
#pragma unroll
    for (int t = 0; t < NT; ++t) acc[t] = v8f{};

    // A fragment (16x4 f32): lanes 0-15 hold K=k..k+1, lanes 16-31 K=k+2..k+3
    const float* __restrict__ arow = A + (size_t)(mBase + l) * lda + 2 * half;
    // B fragment (4x16 f32): same half/K striping, N along lanes
    const float* __restrict__ bcol = B + (size_t)(2 * half) * ldb + nBase + l;

    for (int k = 0; k < K; k += 4) {
        v2f a;
        a.x = arow[k + 0];
        a.y = arow[k + 1];
        const float* __restrict__ bk = bcol + (size_t)k * ldb;
#pragma unroll
        for (int t = 0; t < NT; ++t) {
            v2f b;
            b.x = bk[t * 16];
            b.y = bk[t * 16 + ldb];
            acc[t] = __builtin_amdgcn_wmma_f32_16x16x4_f32(
                false, a, false, b, (short)0, acc[t], false, false);
        }
    }

#pragma unroll
    for (int t = 0; t < NT; ++t) {
        const int n = nBase + t * 16 + l;
#pragma unroll
        for (int r = 0; r < 8; ++r) {
            const int row = mBase + r + 8 * half;
            C[(size_t)row * ldc + n] = acc[t][r];
        }
    }
}

// ---------------- seed scatter destination with bias ----------------
// ld == (1 << logF); pad columns (f >= Fvalid) are zeroed (clears poison).
__global__ void gcn_bias_seed(float* __restrict__ out, const float* __restrict__ b,
                              int total, int logF, int Fvalid) {
    int i = blockIdx.x * blockDim.x + threadIdx.x;
    if (i >= total) return;
    int f = i & ((1 << logF) - 1);
    out[i] = (f < Fvalid) ? b[f] : 0.0f;
}

// ---------------- gather * norm -> scatter-add ----------------
// blockIdx.y = channel group (no integer division).  Edge e >= E is the
// self loop of node (e - E) with norm = dinv^2.
template <int VEC>
__global__ void gcn_aggregate(const float* __restrict__ h,
                              const int* __restrict__ src,
                              const int* __restrict__ dst,
                              const float* __restrict__ dinv,
                              float* __restrict__ out,
                              int E, int Nn, int ldh, int ldo) {
    int e = blockIdx.x * blockDim.x + threadIdx.x;
    if (e >= E + Nn) return;
    int s, d; float nrm;
    if (e < E) { s = src[e]; d = dst[e]; nrm = dinv[s] * dinv[d]; }
    else       { s = d = e - E; float di = dinv[s]; nrm = di * di; }

    const int g = blockIdx.y;
    const float* hp = h   + (size_t)s * ldh + g * VEC;
    float*       op = out + (size_t)d * ldo + g * VEC;

    if (VEC == 4) {
        float4 v = *(const float4*)hp;
        unsafeAtomicAdd(op + 0, v.x * nrm);
        unsafeAtomicAdd(op + 1, v.y * nrm);
        unsafeAtomicAdd(op + 2, v.z * nrm);
        unsafeAtomicAdd(op + 3, v.w * nrm);
    } else {
        float2 v = *(const float2*)hp;
        unsafeAtomicAdd(op + 0, v.x * nrm);
        unsafeAtomicAdd(op + 1, v.y * nrm);
    }
}

// ---------------- ReLU ----------------
__global__ void gcn_relu(float* __restrict__ x, int total) {
    int i = blockIdx.x * blockDim.x + threadIdx.x;
    if (i < total) x[i] = fmaxf(x[i], 0.0f);
}

// ---------------- log_softmax over 10 classes (input ld = 16) ----------------
__global__ void gcn_log_softmax10(const float* __restrict__ in,
                                  float* __restrict__ out, int Nn) {
    int n = blockIdx.x * blockDim.x + threadIdx.x;
    if (n >= Nn) return;
    const float* r = in + (size_t)n * 16;
    float m = r[0];
#pragma unroll
    for (int c = 1; c < 10; ++c) m = fmaxf(m, r[c]);
    float ssum = 0.0f;
#pragma unroll
    for (int c = 0; c < 10; ++c) ssum += __expf(r[c] - m);
    float lse = m + __logf(ssum);
#pragma unroll
    for (int c = 0; c < 10; ++c) out[(size_t)n * 10 + c] = r[c] - lse;
}

// ---------------------------------------------------------------------------
extern "C" void kernel_launch(void* const* d_in, const int* in_sizes, int n_in,
                              void* d_out, int out_size, void* d_ws, size_t ws_size,
                              hipStream_t stream) {
    const float* x  = (const float*)d_in[0];
    const int*   ei = (const int*)d_in[1];     // [2, E] int32 (JAX demotes i64)
    const float* W1 = (const float*)d_in[2];
    const float* b1 = (const float*)d_in[3];
    const float* W2 = (const float*)d_in[4];
    const float* b2 = (const float*)d_in[5];
    const float* W3 = (const float*)d_in[6];
    const float* b3 = (const float*)d_in[7];
    float* out = (float*)d_out;

    const int F_IN = 512, H1 = 256, H2 = 64;
    const int Nn = in_sizes[0] / F_IN;         // 100000
    const int E  = in_sizes[1] / 2;            // 3200000
    const int* srcp = ei;
    const int* dstp = ei + E;

    float* ws    = (float*)d_ws;
    float* dinv  = ws;                                  // N floats
    float* bufA  = ws + Nn;                             // N*256 floats
    float* bufB  = bufA + (size_t)Nn * 256;             // N*256 floats
    float* w3pad = bufB + (size_t)Nn * 256;             // 64*16 floats

    const int T = 256;
    auto cdiv = [](long long a, long long b) { return (int)((a + b - 1) / b); };

    // ---- degrees / symmetric norm + W3 padding ----
    gcn_deg_init<<<cdiv(Nn, T), T, 0, stream>>>(dinv, Nn);
    gcn_deg_count<<<cdiv(E, T), T, 0, stream>>>(dstp, dinv, E);
    gcn_deg_to_dinv<<<cdiv(Nn, T), T, 0, stream>>>(dinv, Nn);
    gcn_pad_w3<<<4, T, 0, stream>>>(W3, w3pad);

    const int mBlocks = cdiv(Nn, 128);
    const int eBlocks = cdiv((long long)E + Nn, T);

    // ---- layer 1: x(512) -> 256 ----
    {
        dim3 g(H1 / 64, mBlocks);
        gcn_gemm_wmma<4><<<g, T, 0, stream>>>(x, W1, bufA, Nn, F_IN, F_IN, H1, H1);
        gcn_bias_seed<<<cdiv((long long)Nn * H1, T), T, 0, stream>>>(bufB, b1, Nn * H1, 8, H1);
        dim3 ga(eBlocks, H1 / 4);
        gcn_aggregate<4><<<ga, T, 0, stream>>>(bufA, srcp, dstp, dinv, bufB, E, Nn, H1, H1);
        gcn_relu<<<cdiv((long long)Nn * H1, T), T, 0, stream>>>(bufB, Nn * H1);
    }

    // ---- layer 2: 256 -> 64 ----
    {
        dim3 g(1, mBlocks);
        gcn_gemm_wmma<4><<<g, T, 0, stream>>>(bufB, W2, bufA, Nn, H1, H1, H2, H2);
        gcn_bias_seed<<<cdiv((long long)Nn * H2, T), T, 0, stream>>>(bufB, b2, Nn * H2, 6, H2);
        dim3 ga(eBlocks, H2 / 4);
        gcn_aggregate<4><<<ga, T, 0, stream>>>(bufA, srcp, dstp, dinv, bufB, E, Nn, H2, H2);
        gcn_relu<<<cdiv((long long)Nn * H2, T), T, 0, stream>>>(bufB, Nn * H2);
    }

    // ---- layer 3: 64 -> 10 (B zero-padded to 64x16, ld 16) + log_softmax ----
    {
        dim3 g(1, mBlocks);
        gcn_gemm_wmma<1><<<g, T, 0, stream>>>(bufB, w3pad, bufA, Nn, H2, H2, 16, 16);
        gcn_bias_seed<<<cdiv((long long)Nn * 16, T), T, 0, stream>>>(bufB, b3, Nn * 16, 4, 10);
        dim3 ga(eBlocks, 5);                       // 10 channels, float2 groups
        gcn_aggregate<2><<<ga, T, 0, stream>>>(bufA, srcp, dstp, dinv, bufB, E, Nn, 16, 16);
        gcn_log_softmax10<<<cdiv(Nn, T), T, 0, stream>>>(bufB, out, Nn);
    }
}